// CrystalGNN_80178449482414
// MI455X (gfx1250) — compile-verified
//
#include <hip/hip_runtime.h>
#include <hip/hip_bf16.h>

typedef __attribute__((ext_vector_type(2))) float v2f;
typedef __attribute__((ext_vector_type(8))) float v8f;

#define F_IN   128
#define HID    64
#define NGRAPH 256

// ---------------- init / degree / norm ----------------

__global__ void k_init_deg(float* __restrict__ deg, int N) {
  int i = blockIdx.x * blockDim.x + threadIdx.x;
  if (i < N) deg[i] = 1.0f;   // self-loop contribution
}

__global__ void k_zero(float* __restrict__ p, int n) {
  int i = blockIdx.x * blockDim.x + threadIdx.x;
  if (i < n) p[i] = 0.0f;
}

__global__ void k_degree(const long long* __restrict__ ei, float* __restrict__ deg, int E) {
  int e = blockIdx.x * blockDim.x + threadIdx.x;
  if (e < E) {
    int c = (int)ei[(size_t)E + e];   // target node
    atomicAdd(&deg[c], 1.0f);
  }
}

__global__ void k_dinv(float* __restrict__ deg, int N) {
  int i = blockIdx.x * blockDim.x + threadIdx.x;
  if (i < N) {
    float d = deg[i];
    deg[i] = (d > 0.0f) ? rsqrtf(d) : 0.0f;
  }
}

// ---------------- h = x @ W1 via V_WMMA_F32_16X16X4_F32 ----------------
// One wave computes a 16-row x 64-col strip. K loop: 128 in steps of 4.
// A (16x4 f32): lane L -> row M=L%16; VGPR0/1 hold K={0,1} (lanes 0-15)
// or K={2,3} (lanes 16-31). B (4x16): lane L -> col N=L%16, same K split.
// C/D: VGPR v -> row v (lanes 0-15) / row v+8 (lanes 16-31), col = L%16.
__global__ void k_gemm(const float* __restrict__ x, const float* __restrict__ W,
                       float* __restrict__ h, int N) {
  const int lane = threadIdx.x & 31;
  const int wave = threadIdx.x >> 5;
  const int tile = blockIdx.x * (blockDim.x >> 5) + wave;   // 16-row tile id
  const int rowBase = tile * 16;
  if (rowBase >= N) return;

  const int m    = lane & 15;
  const int koff = (lane >> 4) << 1;      // 0 or 2
  int arow = rowBase + m;
  if (arow >= N) arow = N - 1;            // safe clamp (N%16==0 in practice)
  const float* __restrict__ xr = x + (size_t)arow * F_IN;

  v8f acc0 = {}, acc1 = {}, acc2 = {}, acc3 = {};

  for (int k = 0; k < F_IN; k += 4) {
    v2f a;
    a[0] = xr[k + koff];
    a[1] = xr[k + koff + 1];

    const float* __restrict__ w0 = W + (size_t)(k + koff) * HID + m;
    const float* __restrict__ w1 = w0 + HID;
    v2f b0, b1, b2, b3;
    b0[0] = w0[0];  b0[1] = w1[0];
    b1[0] = w0[16]; b1[1] = w1[16];
    b2[0] = w0[32]; b2[1] = w1[32];
    b3[0] = w0[48]; b3[1] = w1[48];

    acc0 = __builtin_amdgcn_wmma_f32_16x16x4_f32(false, a, false, b0, (short)0, acc0, false, false);
    acc1 = __builtin_amdgcn_wmma_f32_16x16x4_f32(false, a, false, b1, (short)0, acc1, false, false);
    acc2 = __builtin_amdgcn_wmma_f32_16x16x4_f32(false, a, false, b2, (short)0, acc2, false, false);
    acc3 = __builtin_amdgcn_wmma_f32_16x16x4_f32(false, a, false, b3, (short)0, acc3, false, false);
  }

  const int r0  = rowBase + ((lane < 16) ? 0 : 8);
  const int col = m;
#pragma unroll
  for (int v = 0; v < 8; ++v) {
    int r = r0 + v;
    if (r < N) {
      float* __restrict__ hr = h + (size_t)r * HID + col;
      hr[0]  = acc0[v];
      hr[16] = acc1[v];
      hr[32] = acc2[v];
      hr[48] = acc3[v];
    }
  }
}

// ---------------- edge gather + normalize + scatter-add ----------------
// 16 threads per edge, 4 features each (float4 gather, 4 f32 atomics).
// h and agg both fit in the 192MB L2, so atomics resolve on-chip.
__global__ void k_edge(const long long* __restrict__ ei, const float* __restrict__ dinv,
                       const float* __restrict__ h, float* __restrict__ agg, int E) {
  int tid = blockIdx.x * blockDim.x + threadIdx.x;
  int e = tid >> 4;
  if (e >= E) return;
  int fq = (tid & 15) << 2;
  int r = (int)ei[e];
  int c = (int)ei[(size_t)E + e];
  float nrm = dinv[r] * dinv[c];
  const float4 hv = *(const float4*)(h + (size_t)r * HID + fq);
  float* __restrict__ dst = agg + (size_t)c * HID + fq;
  atomicAdd(dst + 0, nrm * hv.x);
  atomicAdd(dst + 1, nrm * hv.y);
  atomicAdd(dst + 2, nrm * hv.z);
  atomicAdd(dst + 3, nrm * hv.w);
}

// ---------------- fused self-loop + bias + relu + mean-pool accumulate ----------------
__global__ void k_pool(const float* __restrict__ agg, const float* __restrict__ h,
                       const float* __restrict__ dinv, const float* __restrict__ b1,
                       const long long* __restrict__ batch,
                       float* __restrict__ sums, float* __restrict__ cnt, int N) {
  int tid = blockIdx.x * blockDim.x + threadIdx.x;
  int n = tid >> 6;
  if (n >= N) return;
  int f = tid & 63;
  float di = dinv[n];
  size_t off = (size_t)n * HID + f;
  float val = agg[off] + di * di * h[off] + b1[f];   // self-loop fused here
  val = fmaxf(val, 0.0f);                            // relu
  int g = (int)batch[n];
  atomicAdd(&sums[g * HID + f], val);
  if (f == 0) atomicAdd(&cnt[g], 1.0f);
}

// ---------------- mean, classifier, log-softmax ----------------
__global__ void k_final(const float* __restrict__ sums, const float* __restrict__ cnt,
                        const float* __restrict__ Wfc, const float* __restrict__ bfc,
                        float* __restrict__ out) {
  int g = threadIdx.x;          // one block of NGRAPH threads
  float c = fmaxf(cnt[g], 1.0f);
  float inv = 1.0f / c;
  float l0 = bfc[0], l1 = bfc[1];
#pragma unroll 4
  for (int k = 0; k < HID; ++k) {
    float p = sums[g * HID + k] * inv;
    l0 += p * Wfc[k * 2 + 0];
    l1 += p * Wfc[k * 2 + 1];
  }
  float mx  = fmaxf(l0, l1);
  float lse = mx + logf(expf(l0 - mx) + expf(l1 - mx));
  out[g * 2 + 0] = l0 - lse;
  out[g * 2 + 1] = l1 - lse;
}

extern "C" void kernel_launch(void* const* d_in, const int* in_sizes, int n_in,
                              void* d_out, int out_size, void* d_ws, size_t ws_size,
                              hipStream_t stream) {
  (void)n_in; (void)out_size; (void)ws_size;
  const float*     x     = (const float*)d_in[0];
  const long long* ei    = (const long long*)d_in[1];
  const long long* batch = (const long long*)d_in[2];
  const float*     W1    = (const float*)d_in[3];
  const float*     b1    = (const float*)d_in[4];
  const float*     Wfc   = (const float*)d_in[5];
  const float*     bfc   = (const float*)d_in[6];
  float*           out   = (float*)d_out;

  const int N = in_sizes[2];          // 100000
  const int E = in_sizes[1] / 2;      // 1600000

  // workspace layout (floats): dinv | h | agg | sums | cnt  (agg..cnt contiguous)
  float* ws   = (float*)d_ws;
  float* dinv = ws;
  float* h    = dinv + N;
  float* agg  = h + (size_t)N * HID;
  float* sums = agg + (size_t)N * HID;
  float* cnt  = sums + NGRAPH * HID;

  const int zcount = N * HID + NGRAPH * HID + NGRAPH;   // zero agg+sums+cnt in one pass

  k_init_deg<<<(N + 255) / 256, 256, 0, stream>>>(dinv, N);
  k_zero<<<(zcount + 255) / 256, 256, 0, stream>>>(agg, zcount);
  k_degree<<<(E + 255) / 256, 256, 0, stream>>>(ei, dinv, E);
  k_dinv<<<(N + 255) / 256, 256, 0, stream>>>(dinv, N);

  const int tiles = (N + 15) / 16;                       // 16-row tiles
  k_gemm<<<(tiles + 7) / 8, 256, 0, stream>>>(x, W1, h, N);

  const long long et = (long long)E * 16;
  k_edge<<<(unsigned)((et + 255) / 256), 256, 0, stream>>>(ei, dinv, h, agg, E);

  const long long pt = (long long)N * HID;
  k_pool<<<(unsigned)((pt + 255) / 256), 256, 0, stream>>>(agg, h, dinv, b1, batch, sums, cnt, N);

  k_final<<<1, NGRAPH, 0, stream>>>(sums, cnt, Wfc, bfc, out);
}